// one_gnn_15633680957442
// MI455X (gfx1250) — compile-verified
//
#include <hip/hip_runtime.h>
#include <math.h>

// Problem constants (match reference)
#define NNODES 10000
#define NEDGES 640000
#define NPAIRS 500000
#define DIN    128
#define DHID   256
#define DEMB   64

typedef __attribute__((ext_vector_type(2))) float v2f;
typedef __attribute__((ext_vector_type(8))) float v8f;

// ---------------- degree / norm ----------------
__global__ void degree_kernel(const int* __restrict__ src, const int* __restrict__ dst,
                              float* __restrict__ deg_out, float* __restrict__ deg_in) {
    int e = blockIdx.x * blockDim.x + threadIdx.x;
    if (e < NEDGES) {
        unsafeAtomicAdd(&deg_out[src[e]], 1.0f);
        unsafeAtomicAdd(&deg_in[dst[e]], 1.0f);
    }
}

// in-place: deg -> norm = deg>0 ? rsqrt(max(deg,1)) : 0
__global__ void norm_kernel(float* __restrict__ deg_out, float* __restrict__ deg_in) {
    int i = blockIdx.x * blockDim.x + threadIdx.x;
    if (i < NNODES) {
        float o = deg_out[i], n = deg_in[i];
        deg_out[i] = (o > 0.0f) ? rsqrtf(fmaxf(o, 1.0f)) : 0.0f;
        deg_in[i]  = (n > 0.0f) ? rsqrtf(fmaxf(n, 1.0f)) : 0.0f;
    }
}

// ---------------- prescale: hs = h * norm_out[row]  (float4 per thread) ----------------
template<int D>
__global__ void prescale_kernel(const float* __restrict__ h, const float* __restrict__ norm_out,
                                float* __restrict__ hs) {
    constexpr int C  = D / 4;                 // float4 chunks per row
    constexpr int SH = (D == 128) ? 5 : 6;    // log2(C)
    int i = blockIdx.x * blockDim.x + threadIdx.x;
    if (i < NNODES * C) {
        int row = i >> SH;
        float s = norm_out[row];
        float4 v = ((const float4*)h)[i];
        v.x *= s; v.y *= s; v.z *= s; v.w *= s;
        ((float4*)hs)[i] = v;
    }
}

// ---------------- edge scatter-add: agg[dst] += hs[src]  (float4 per thread) ----------------
// L2-resident src/dst tensors; native global_atomic_add_f32 via unsafeAtomicAdd.
template<int D>
__global__ void edge_agg_kernel(const int* __restrict__ src, const int* __restrict__ dst,
                                const float* __restrict__ hs, float* __restrict__ agg) {
    constexpr int C  = D / 4;
    constexpr int SH = (D == 128) ? 5 : 6;    // log2(C)
    int i = blockIdx.x * blockDim.x + threadIdx.x;   // <= 40.96M, fits int32
    if (i < NEDGES * C) {
        int e = i >> SH;
        int c = (i & (C - 1)) << 2;
        int s = src[e], t = dst[e];
        const float4 v = *(const float4*)(hs + (size_t)s * D + c);
        float* a = agg + (size_t)t * D + c;
        unsafeAtomicAdd(a + 0, v.x);
        unsafeAtomicAdd(a + 1, v.y);
        unsafeAtomicAdd(a + 2, v.z);
        unsafeAtomicAdd(a + 3, v.w);
    }
}

// ---------------- WMMA GEMM: C = act( diag(norm_in) * (A @ W) + bias ) ----------------
// A: [M=10000, K] row-major (aggregated features; norm_in row-scaling folded into epilogue,
// valid since diag(s)*A*W == diag(s)*(A*W)).  W: [K, NOUT] row-major.
// One wave computes a 16x64 strip: 4x v_wmma_f32_16x16x4_f32 per K-step, fp32 in/accum
// (GEMM flops are negligible here, so no reason to drop precision to bf16).
// ACT: 0 = relu, 1 = sigmoid
template<int K, int NOUT, int ACT>
__global__ void gemm_wmma_kernel(const float* __restrict__ A, const float* __restrict__ W,
                                 const float* __restrict__ bias, const float* __restrict__ norm_in,
                                 float* __restrict__ C, int total_waves) {
    const int lane = threadIdx.x & 31;
    const int wid  = blockIdx.x * (blockDim.x >> 5) + (threadIdx.x >> 5);
    if (wid >= total_waves) return;           // wave-uniform: EXEC stays all-1s for WMMA

    constexpr int NGRP = NOUT / 64;           // 64 output columns per wave
    const int m0  = (wid / NGRP) * 16;
    const int n0  = (wid % NGRP) * 64;
    const int col = lane & 15;
    const int kg  = (lane >> 4) << 1;         // K sub-group per lane half: 0 or 2

    const float* arow = A + (long long)(m0 + col) * K;   // A row for this lane

    v8f acc0 = {}, acc1 = {}, acc2 = {}, acc3 = {};
    #pragma unroll 2
    for (int k = 0; k < K; k += 4) {
        // A 16x4 fp32 tile: lane<16 holds K={k,k+1}, lane>=16 holds K={k+2,k+3}
        v2f a = *(const v2f*)(arow + k + kg);
        const float* w0 = W + (long long)(k + kg) * NOUT + n0 + col;
        const float* w1 = w0 + NOUT;
        v2f b0 = { w0[0],  w1[0]  };
        v2f b1 = { w0[16], w1[16] };
        v2f b2 = { w0[32], w1[32] };
        v2f b3 = { w0[48], w1[48] };
        acc0 = __builtin_amdgcn_wmma_f32_16x16x4_f32(false, a, false, b0, (short)0, acc0, false, false);
        acc1 = __builtin_amdgcn_wmma_f32_16x16x4_f32(false, a, false, b1, (short)0, acc1, false, false);
        acc2 = __builtin_amdgcn_wmma_f32_16x16x4_f32(false, a, false, b2, (short)0, acc2, false, false);
        acc3 = __builtin_amdgcn_wmma_f32_16x16x4_f32(false, a, false, b3, (short)0, acc3, false, false);
    }

    // epilogue: row scale by norm_in, add bias, activation
    const int rbase = m0 + ((lane >> 4) << 3);   // lanes 16-31 hold rows m0+8..m0+15
    const float bv0 = bias[n0 + col];
    const float bv1 = bias[n0 + 16 + col];
    const float bv2 = bias[n0 + 32 + col];
    const float bv3 = bias[n0 + 48 + col];
    #pragma unroll
    for (int r = 0; r < 8; ++r) {
        const int row = rbase + r;
        const float s = norm_in[row];
        float v0 = acc0[r] * s + bv0;
        float v1 = acc1[r] * s + bv1;
        float v2 = acc2[r] * s + bv2;
        float v3 = acc3[r] * s + bv3;
        if (ACT == 0) {
            v0 = fmaxf(v0, 0.0f); v1 = fmaxf(v1, 0.0f);
            v2 = fmaxf(v2, 0.0f); v3 = fmaxf(v3, 0.0f);
        } else {
            v0 = 1.0f / (1.0f + expf(-v0)); v1 = 1.0f / (1.0f + expf(-v1));
            v2 = 1.0f / (1.0f + expf(-v2)); v3 = 1.0f / (1.0f + expf(-v3));
        }
        float* crow = C + (long long)row * NOUT + n0 + col;
        crow[0]  = v0;
        crow[16] = v1;
        crow[32] = v2;
        crow[48] = v3;
    }
}

// ---------------- head: out[p] = tanh([o_emb, d_emb, dis] @ Wout + bout) ----------------
__global__ void pair_kernel(const float* __restrict__ emb, const float* __restrict__ dis,
                            const float* __restrict__ Wout, const float* __restrict__ bout,
                            const int* __restrict__ tidx, float* __restrict__ out) {
    int p = blockIdx.x * blockDim.x + threadIdx.x;
    if (p >= NPAIRS) return;
    const int o = tidx[p];
    const int d = tidx[NPAIRS + p];
    const float4* oe = (const float4*)(emb + (long long)o * DEMB);
    const float4* de = (const float4*)(emb + (long long)d * DEMB);
    const float4* wo = (const float4*)(Wout);
    const float4* wd = (const float4*)(Wout + DEMB);
    float acc = bout[0];
    #pragma unroll
    for (int j = 0; j < DEMB / 4; ++j) {
        float4 a = oe[j], wa = wo[j];
        float4 b = de[j], wb = wd[j];
        acc += a.x * wa.x + a.y * wa.y + a.z * wa.z + a.w * wa.w;
        acc += b.x * wb.x + b.y * wb.y + b.z * wb.z + b.w * wb.w;
    }
    acc += dis[(long long)o * NNODES + d] * Wout[2 * DEMB];
    out[p] = tanhf(acc);
}

extern "C" void kernel_launch(void* const* d_in, const int* in_sizes, int n_in,
                              void* d_out, int out_size, void* d_ws, size_t ws_size,
                              hipStream_t stream) {
    (void)in_sizes; (void)n_in; (void)out_size; (void)ws_size;
    const float* x    = (const float*)d_in[0];
    const float* dis  = (const float*)d_in[1];
    const float* W1   = (const float*)d_in[2];
    const float* b1   = (const float*)d_in[3];
    const float* W2   = (const float*)d_in[4];
    const float* b2   = (const float*)d_in[5];
    const float* W3   = (const float*)d_in[6];
    const float* b3   = (const float*)d_in[7];
    const float* Wout = (const float*)d_in[8];
    const float* bout = (const float*)d_in[9];
    const int*   src  = (const int*)d_in[10];
    const int*   dst  = (const int*)d_in[11];
    const int*   tidx = (const int*)d_in[12];
    float* out = (float*)d_out;

    // workspace carve-out (256B aligned)
    char* ws = (char*)d_ws;
    size_t off = 0;
    auto carve = [&](size_t bytes) -> float* {
        float* p = (float*)(ws + off);
        off = (off + bytes + 255) & ~(size_t)255;
        return p;
    };
    float* norm_out = carve((size_t)NNODES * 4);          // deg_out -> norm_out in-place
    float* norm_in  = carve((size_t)NNODES * 4);          // deg_in  -> norm_in  in-place
    float* hs       = carve((size_t)NNODES * DHID * 4);   // prescaled features
    float* agg      = carve((size_t)NNODES * DHID * 4);   // aggregated features
    float* h        = carve((size_t)NNODES * DHID * 4);   // hidden activations
    float* gemb     = carve((size_t)NNODES * DEMB * 4);   // final node embeddings

    // degrees + norms
    hipMemsetAsync(norm_out, 0, (size_t)NNODES * 4, stream);
    hipMemsetAsync(norm_in,  0, (size_t)NNODES * 4, stream);
    degree_kernel<<<(NEDGES + 255) / 256, 256, 0, stream>>>(src, dst, norm_out, norm_in);
    norm_kernel<<<(NNODES + 255) / 256, 256, 0, stream>>>(norm_out, norm_in);

    // layer 1: x[10000,128] -> h[10000,256], relu
    hipMemsetAsync(agg, 0, (size_t)NNODES * DIN * 4, stream);
    prescale_kernel<DIN><<<(NNODES * (DIN / 4) + 255) / 256, 256, 0, stream>>>(x, norm_out, hs);
    edge_agg_kernel<DIN><<<(NEDGES * (DIN / 4)) / 256, 256, 0, stream>>>(src, dst, hs, agg);
    {   // 625 M-tiles x 4 col-groups = 2500 waves; 4 waves/block
        int waves = (NNODES / 16) * (DHID / 64);
        gemm_wmma_kernel<DIN, DHID, 0><<<(waves + 3) / 4, 128, 0, stream>>>(agg, W1, b1, norm_in, h, waves);
    }

    // layer 2: h[10000,256] -> h[10000,256], relu
    hipMemsetAsync(agg, 0, (size_t)NNODES * DHID * 4, stream);
    prescale_kernel<DHID><<<(NNODES * (DHID / 4) + 255) / 256, 256, 0, stream>>>(h, norm_out, hs);
    edge_agg_kernel<DHID><<<(NEDGES * (DHID / 4)) / 256, 256, 0, stream>>>(src, dst, hs, agg);
    {
        int waves = (NNODES / 16) * (DHID / 64);
        gemm_wmma_kernel<DHID, DHID, 0><<<(waves + 3) / 4, 128, 0, stream>>>(agg, W2, b2, norm_in, h, waves);
    }

    // layer 3: h[10000,256] -> gemb[10000,64], sigmoid
    hipMemsetAsync(agg, 0, (size_t)NNODES * DHID * 4, stream);
    prescale_kernel<DHID><<<(NNODES * (DHID / 4) + 255) / 256, 256, 0, stream>>>(h, norm_out, hs);
    edge_agg_kernel<DHID><<<(NEDGES * (DHID / 4)) / 256, 256, 0, stream>>>(src, dst, hs, agg);
    {
        int waves = (NNODES / 16) * (DEMB / 64);   // 625 waves
        gemm_wmma_kernel<DHID, DEMB, 1><<<(waves + 3) / 4, 128, 0, stream>>>(agg, W3, b3, norm_in, gemb, waves);
    }

    // head
    pair_kernel<<<(NPAIRS + 255) / 256, 256, 0, stream>>>(gemb, dis, Wout, bout, tidx, out);
}